// Router_44074954392149
// MI455X (gfx1250) — compile-verified
//
#include <hip/hip_runtime.h>
#include <hip/hip_bf16.h>
#include <math.h>

// MoE router, fused single pass over x.
//   logits[t,e]      = x[t,:] . route_w[e,:] + route_b[e]
//   noise_log[t,e]   = x[t,:] . noise_w[e,:] + noise_b[e]
//   noisy[t,e]       = logits + noise[t,e]*softplus(noise_log)
//   top2 -> sparse softmax -> router_output [T,8], indices [T,2]
//
// GEMM view: M=T tokens, N=16 (8 route | 8 noise), K=2048, f32.
// Matrix op: V_WMMA_F32_16X16X4_F32 (exact f32, matches reference numerics).
// One wave = one 16-token tile; 512 wmma ops per tile; unroll 8 so 16
// global_load_b64 are in flight per wave (latency hiding for HBM roof).

#define N_EMBED  2048
#define N_EXP    8
#define TILE_M   16
#define WAVES_PER_BLOCK 2
#define THREADS_PER_BLOCK (WAVES_PER_BLOCK * 32)

typedef __attribute__((ext_vector_type(2))) float v2f;
typedef __attribute__((ext_vector_type(8))) float v8f;

__device__ __forceinline__ float softplus_stable(float z) {
    // log1p(exp(z)), stable for large |z|
    return (z > 0.0f) ? (z + log1pf(__expf(-z))) : log1pf(__expf(z));
}

__global__ __launch_bounds__(THREADS_PER_BLOCK)
void router_fused_kernel(const float* __restrict__ x,
                         const float* __restrict__ noise,
                         const float* __restrict__ route_w,
                         const float* __restrict__ route_b,
                         const float* __restrict__ noise_w,
                         const float* __restrict__ noise_b,
                         float* __restrict__ out,       // [T, 8]
                         int*   __restrict__ idx_out,   // [T, 2]
                         int T) {
    __shared__ float tile[WAVES_PER_BLOCK][TILE_M][16];

    const int tid     = blockIdx.x * blockDim.x + threadIdx.x;
    const int wave    = tid >> 5;                 // global wave id = tile id
    const int wlocal  = threadIdx.x >> 5;         // wave within block
    const int lane    = threadIdx.x & 31;
    const int n_tiles = T / TILE_M;
    if (wave >= n_tiles) return;                  // wave-uniform branch

    const int tok_base = wave * TILE_M;

    // ---- fragment address setup (ISA 7.12.2 layouts) ----
    const int lo   = lane & 15;       // row (A) / col (B) index 0..15
    const int koff = (lane >> 4) * 2; // lo lanes K+0,K+1 ; hi lanes K+2,K+3

    // A: x row = token (tok_base + lo), columns koff..koff+1 of each K chunk
    const float* aptr = x + (size_t)(tok_base + lo) * N_EMBED + koff;
    // B: col 0..7 -> route_w rows, col 8..15 -> noise_w rows
    const float* wrow = (lo < N_EXP) ? (route_w + (size_t)lo * N_EMBED)
                                     : (noise_w + (size_t)(lo - N_EXP) * N_EMBED);
    const float* bptr = wrow + koff;

    v8f acc = {};

    // ---- K loop: 2048 / (4 per wmma) = 512 wmma, unrolled 8 ----
    for (int k = 0; k < N_EMBED; k += 32) {
        v2f a[8], b[8];
#pragma unroll
        for (int u = 0; u < 8; ++u) {
            a[u] = *(const v2f*)(aptr + k + u * 4);
            b[u] = *(const v2f*)(bptr + k + u * 4);
        }
#pragma unroll
        for (int u = 0; u < 8; ++u) {
            acc = __builtin_amdgcn_wmma_f32_16x16x4_f32(
                /*neg_a=*/false, a[u], /*neg_b=*/false, b[u],
                /*c_mod=*/(short)0, acc, /*reuse_a=*/false, /*reuse_b=*/false);
        }
    }

    // ---- spill D tile to LDS: acc[r] holds (M = r + 8*(lane>=16), N = lo) ----
    const int mhi = (lane >> 4) * 8;
#pragma unroll
    for (int r = 0; r < 8; ++r) {
        tile[wlocal][r + mhi][lo] = acc[r];
    }
    __syncthreads();

    // ---- epilogue: 16 lanes, one token each ----
    if (lane < TILE_M) {
        const int tok = tok_base + lane;
        float nz[N_EXP];
#pragma unroll
        for (int e = 0; e < N_EXP; ++e) {
            float lg = tile[wlocal][lane][e] + route_b[e];
            float z  = tile[wlocal][lane][e + 8] + noise_b[e];
            nz[e] = lg + noise[(size_t)tok * N_EXP + e] * softplus_stable(z);
        }
        // top-2, strict '>' keeps lowest index on ties (jax top_k semantics)
        int i1 = 0; float v1 = nz[0];
#pragma unroll
        for (int e = 1; e < N_EXP; ++e)
            if (nz[e] > v1) { v1 = nz[e]; i1 = e; }
        int i2 = -1; float v2 = -INFINITY;
#pragma unroll
        for (int e = 0; e < N_EXP; ++e)
            if (e != i1 && nz[e] > v2) { v2 = nz[e]; i2 = e; }

        // softmax over {v1, v2}; all other entries exp(-inf) = 0
        const float ex = __expf(v2 - v1);
        const float inv = 1.0f / (1.0f + ex);
        float o[N_EXP];
#pragma unroll
        for (int e = 0; e < N_EXP; ++e) o[e] = 0.0f;
        o[i1] = inv;
        o[i2] = ex * inv;

#pragma unroll
        for (int e = 0; e < N_EXP; ++e)
            out[(size_t)tok * N_EXP + e] = o[e];
        idx_out[(size_t)tok * 2 + 0] = i1;
        idx_out[(size_t)tok * 2 + 1] = i2;
    }
}

extern "C" void kernel_launch(void* const* d_in, const int* in_sizes, int n_in,
                              void* d_out, int out_size, void* d_ws, size_t ws_size,
                              hipStream_t stream) {
    const float* x       = (const float*)d_in[0];
    const float* noise   = (const float*)d_in[1];
    const float* route_w = (const float*)d_in[2];
    const float* route_b = (const float*)d_in[3];
    const float* noise_w = (const float*)d_in[4];
    const float* noise_b = (const float*)d_in[5];

    const int T = in_sizes[0] / N_EMBED;          // 16384 tokens
    float* out   = (float*)d_out;                 // [T,8] router_output
    int*   idx   = (int*)(out + (size_t)T * N_EXP); // [T,2] indices (int32 region)

    const int n_tiles = T / TILE_M;               // 1024 wave-tiles
    dim3 block(THREADS_PER_BLOCK);
    dim3 grid((n_tiles + WAVES_PER_BLOCK - 1) / WAVES_PER_BLOCK);

    router_fused_kernel<<<grid, block, 0, stream>>>(
        x, noise, route_w, route_b, noise_w, noise_b, out, idx, T);
}